// GraphSAGELayer_26044681683126
// MI455X (gfx1250) — compile-verified
//
#include <hip/hip_runtime.h>

typedef __attribute__((ext_vector_type(2))) float v2f;
typedef __attribute__((ext_vector_type(8))) float v8f;

#define LN_EPS 1e-5f
#define F_IN 128
#define F_OUT 128
#define K_TOT 256
// LDS copy of W: 128 rows x 256 cols, row stride padded to 260 floats.
// 260*4B = 1040B row pitch -> 16B aligned (legal B128 stores), and
// 260 mod 64 = 4 banks/row -> lanes 0-15 use banks {4i,4i+1}, lanes 16-31
// (K offset +2) use banks {4i+2,4i+3}: conflict-free b64 fragment reads.
#define W_STRIDE 260
#define W_LDS_BYTES (W_STRIDE * F_OUT * 4)

// ---------------------------------------------------------------------------
// Kernel 1: zero the workspace (msg_sum [N*128] + deg [N]) with float4 stores.
// ---------------------------------------------------------------------------
__global__ __launch_bounds__(256) void zero_ws_kernel(float* __restrict__ p, size_t n) {
    size_t i4 = ((size_t)blockIdx.x * blockDim.x + threadIdx.x) * 4;
    if (i4 + 3 < n) {
        *(float4*)(p + i4) = make_float4(0.f, 0.f, 0.f, 0.f);
    } else {
        for (size_t j = i4; j < n; ++j) p[j] = 0.0f;
    }
}

// ---------------------------------------------------------------------------
// Kernel 2: edge scatter. One wave32 per edge: lane l loads float4 of h[src]
// at column 4l (coalesced 512B per wave) and does 4 f32 atomic adds into
// msg_sum[dst]. Lane 0 bumps deg[dst]. msg_sum (51MB) + h (51MB) both fit in
// the 192MB L2, so gathers and atomics are L2-resident.
// ---------------------------------------------------------------------------
__global__ __launch_bounds__(256) void edge_scatter_kernel(
    const float* __restrict__ h, const int* __restrict__ src,
    const int* __restrict__ dst, float* __restrict__ msg,
    float* __restrict__ deg, int E) {
    int edge = (int)(((size_t)blockIdx.x * blockDim.x + threadIdx.x) >> 5);
    int lane = threadIdx.x & 31;
    if (edge >= E) return;
    int s = src[edge];                     // wave-uniform -> scalar load
    int d = dst[edge];
    float4 hv = ((const float4*)(h + (size_t)s * F_IN))[lane];
    float* mrow = msg + (size_t)d * F_IN + lane * 4;
    atomicAdd(mrow + 0, hv.x);
    atomicAdd(mrow + 1, hv.y);
    atomicAdd(mrow + 2, hv.z);
    atomicAdd(mrow + 3, hv.w);
    if (lane == 0) atomicAdd(deg + d, 1.0f);
}

// ---------------------------------------------------------------------------
// Kernel 3: fused GEMM (f32 WMMA 16x16x4) + mean-div + bias + LayerNorm + ReLU.
// W is staged once per block into padded LDS (coalesced B128 fill), then the
// hot loop does 1 global b64 (A fragment) + 8 conflict-free ds b64 (B
// fragments) per 8 WMMAs. One wave owns a 16-row x 128-col output slab
// (8 v8f accumulators).
// A fragment (16x4 f32): lanes 0-15 hold K={k,k+1}, lanes 16-31 K={k+2,k+3},
// row n0+(lane&15). B fragment (4x16): lane holds W[o0+(lane&15)][kk..kk+1].
// C/D layout: VGPR i -> row i (lanes 0-15) / i+8 (lanes 16-31), col = lane&15.
// ---------------------------------------------------------------------------
__global__ __launch_bounds__(128) void gemm_ln_relu_kernel(
    const float* __restrict__ h, const float* __restrict__ msg,
    const float* __restrict__ deg, const float* __restrict__ W,
    const float* __restrict__ bias, const float* __restrict__ gamma,
    const float* __restrict__ beta, float* __restrict__ out, int N) {
    extern __shared__ float wsm[];  // [F_OUT][W_STRIDE]

    // --- Stage W into LDS: 64 consecutive threads copy one 1KB row ---
    for (int idx = threadIdx.x; idx < F_OUT * (K_TOT / 4); idx += 128) {
        int row = idx >> 6;            // 0..127
        int c   = (idx & 63) << 2;     // float col 0,4,...,252
        *(float4*)(wsm + (size_t)row * W_STRIDE + c) =
            *(const float4*)(W + (size_t)row * K_TOT + c);
    }
    __syncthreads();

    const int lane = threadIdx.x & 31;
    const int waveInBlk = threadIdx.x >> 5;
    const int tile = blockIdx.x * 4 + waveInBlk;  // 16-row tile index
    const int n0 = tile * 16;
    if (n0 < N) {                                 // wave-uniform (EXEC all-1 for WMMA)
        const int l15 = lane & 15;
        const int hi  = lane >> 4;                // 0 -> K offset {0,1}; 1 -> {2,3}
        const int arow = n0 + l15;
        const float rdeg = 1.0f / fmaxf(deg[arow], 1.0f);
        const float* __restrict__ hrow = h   + (size_t)arow * F_IN;
        const float* __restrict__ mrow = msg + (size_t)arow * F_IN;
        const float* __restrict__ wlane = wsm + (size_t)l15 * W_STRIDE;

        v8f c[8];
#pragma unroll
        for (int t = 0; t < 8; ++t) c[t] = v8f{};

        // --- K-half 1: self features h ---
#pragma unroll 2
        for (int k = 0; k < F_IN; k += 4) {
            const int kk = k + hi * 2;
            v2f a = *(const v2f*)(hrow + kk);
#pragma unroll
            for (int t = 0; t < 8; ++t) {
                v2f bf = *(const v2f*)(wlane + (size_t)t * 16 * W_STRIDE + kk);
                c[t] = __builtin_amdgcn_wmma_f32_16x16x4_f32(
                    false, a, false, bf, (short)0, c[t], false, false);
            }
        }
        // --- K-half 2: neighbor mean, msg_sum * (1/max(deg,1)) fused into A ---
#pragma unroll 2
        for (int k = 0; k < F_IN; k += 4) {
            const int kk = k + hi * 2;
            v2f a = *(const v2f*)(mrow + kk);
            a.x *= rdeg;
            a.y *= rdeg;
#pragma unroll
            for (int t = 0; t < 8; ++t) {
                v2f bf = *(const v2f*)(wlane + (size_t)t * 16 * W_STRIDE + F_IN + kk);
                c[t] = __builtin_amdgcn_wmma_f32_16x16x4_f32(
                    false, a, false, bf, (short)0, c[t], false, false);
            }
        }

        // --- Epilogue: bias, LayerNorm over 128 cols per row, affine, ReLU ---
        float bv[8], gv[8], ev[8];
#pragma unroll
        for (int t = 0; t < 8; ++t) {
            int col = t * 16 + l15;
            bv[t] = bias[col];
            gv[t] = gamma[col];
            ev[t] = beta[col];
        }
#pragma unroll
        for (int t = 0; t < 8; ++t)
#pragma unroll
            for (int i = 0; i < 8; ++i) c[t][i] += bv[t];

        const float inv_f = 1.0f / (float)F_OUT;
#pragma unroll
        for (int i = 0; i < 8; ++i) {
            // row r = n0 + i + hi*8; its 128 values live in this 16-lane half,
            // element i of accumulators c[0..7].
            float s = 0.0f, q = 0.0f;
#pragma unroll
            for (int t = 0; t < 8; ++t) {
                float z = c[t][i];
                s += z;
                q += z * z;
            }
            // butterfly within 16-lane halves (wave32; masks 1,2,4,8 stay in-half)
#pragma unroll
            for (int m = 1; m <= 8; m <<= 1) {
                s += __shfl_xor(s, m, 32);
                q += __shfl_xor(q, m, 32);
            }
            float mean = s * inv_f;
            float var  = q * inv_f - mean * mean;
            float rstd = rsqrtf(var + LN_EPS);
            float* orow = out + (size_t)(n0 + i + hi * 8) * F_OUT;
#pragma unroll
            for (int t = 0; t < 8; ++t) {
                float z = (c[t][i] - mean) * rstd * gv[t] + ev[t];
                orow[t * 16 + l15] = fmaxf(z, 0.0f);
            }
        }
    }
}

// ---------------------------------------------------------------------------
extern "C" void kernel_launch(void* const* d_in, const int* in_sizes, int n_in,
                              void* d_out, int out_size, void* d_ws, size_t ws_size,
                              hipStream_t stream) {
    const float* h     = (const float*)d_in[0];
    const int*   src   = (const int*)d_in[1];
    const int*   dst   = (const int*)d_in[2];
    const float* W     = (const float*)d_in[3];
    const float* bias  = (const float*)d_in[4];
    const float* gamma = (const float*)d_in[5];
    const float* beta  = (const float*)d_in[6];
    float* out = (float*)d_out;

    const int N = in_sizes[0] / F_IN;
    const int E = in_sizes[1];

    float* msg = (float*)d_ws;                  // [N * 128]
    float* deg = msg + (size_t)N * F_IN;        // [N]

    // 1) zero msg_sum + deg
    size_t nz = (size_t)N * (F_IN + 1);
    int zblocks = (int)((nz / 4 + 255) / 256) + 1;
    zero_ws_kernel<<<zblocks, 256, 0, stream>>>(msg, nz);

    // 2) edge scatter: 1 wave per edge, 8 edges per 256-thread block
    int eblocks = (E + 7) / 8;
    edge_scatter_kernel<<<eblocks, 256, 0, stream>>>(h, src, dst, msg, deg, E);

    // 3) fused WMMA GEMM + LN + ReLU: 1 wave per 16-row tile, 4 waves per
    //    block, W staged in 130KB of LDS (2 blocks per 320KB WGP).
    int tiles = (N + 15) / 16;
    int gblocks = (tiles + 3) / 4;
    gemm_ln_relu_kernel<<<gblocks, 128, W_LDS_BYTES, stream>>>(
        h, msg, deg, W, bias, gamma, beta, out, N);
}